// Echo_37409165148563
// MI455X (gfx1250) — compile-verified
//
#include <hip/hip_runtime.h>
#include <math.h>

// ============================================================================
// Echo-state network, restructured for MI455X (gfx1250):
//   R_eff = r_r + o_r^T W   (folds output feedback into the recurrence)
//   c     = b @ o_r
//   U     = input @ i_r                       [WMMA f32 GEMM]
//   loop: state(t) = tanh(U[t] + R_eff@state(t-1) + c)   (t=0: tanh(U[0]))
//         (persistent kernel, R_eff rows LDS-resident, 1 grid barrier/step)
//   Out   = S @ W^T + b                       [WMMA f32 GEMM]
// Workspace use: U 64MB + S 64MB + R_eff 16MB + small vectors (~144 MB).
// ============================================================================

typedef __attribute__((ext_vector_type(2))) float v2f;
typedef __attribute__((ext_vector_type(8))) float v8f;

#define T_STEPS 8192
#define IN_DIM  128
#define OUT_DIM 128
#define RSIZE   2048
#define NB      128          // persistent blocks (one 128KB-LDS block per WGP)
#define ROWS_PB 16           // reservoir rows owned per persistent block

__device__ __forceinline__ v8f wmma_f32_16x16x4(v2f a, v2f b, v8f c) {
  // D = A(16x4, f32) * B(4x16, f32) + C(16x16, f32) -- full fp32 precision
  return __builtin_amdgcn_wmma_f32_16x16x4_f32(
      /*neg_a=*/false, a, /*neg_b=*/false, b,
      /*c_mod=*/(short)0, c, /*reuse_a=*/false, /*reuse_b=*/false);
}

// ---------------------------------------------------------------------------
__global__ void init_barrier_kernel(unsigned* bar) { bar[0] = 0u; bar[1] = 0u; }

// c[j] = sum_k b[k] * o_r[k, j]
__global__ void c_vec_kernel(const float* __restrict__ o_r,
                             const float* __restrict__ b,
                             float* __restrict__ cvec) {
  int j = blockIdx.x * blockDim.x + threadIdx.x;
  if (j < RSIZE) {
    float s = 0.f;
#pragma unroll 4
    for (int k = 0; k < OUT_DIM; ++k) s = fmaf(b[k], o_r[k * RSIZE + j], s);
    cvec[j] = s;
  }
}

// ---------------------------------------------------------------------------
// U[T, RSIZE] = X[T, IN] @ i_r[IN, RSIZE]     (512 x 128 tiles of 16x16)
__global__ void gemm_u_kernel(const float* __restrict__ X,
                              const float* __restrict__ i_r,
                              float* __restrict__ U) {
  const int lane = threadIdx.x & 31;
  const int wave = threadIdx.x >> 5;
  const int tile = blockIdx.x * 8 + wave;
  const int tn = tile & (RSIZE / 16 - 1);   // 0..127
  const int tm = tile >> 7;                 // 0..511
  const int l16 = lane & 15, hi = lane >> 4;
  const int m = tm * 16 + l16;              // A row
  const int n = tn * 16 + l16;              // B col
  v8f acc = {};
  for (int kb = 0; kb < IN_DIM; kb += 4) {
    const int ka = kb + 2 * hi;
    v2f a, bb;
    a.x = X[m * IN_DIM + ka];
    a.y = X[m * IN_DIM + ka + 1];
    bb.x = i_r[ka * RSIZE + n];
    bb.y = i_r[(ka + 1) * RSIZE + n];
    acc = wmma_f32_16x16x4(a, bb, acc);
  }
#pragma unroll
  for (int r = 0; r < 8; ++r)
    U[(size_t)(tm * 16 + r + 8 * hi) * RSIZE + n] = acc[r];
}

// ---------------------------------------------------------------------------
// R_eff[j,i] = r_r[j,i] + sum_k o_r[k,j] * W[k,i]     (128 x 128 tiles)
__global__ void gemm_reff_kernel(const float* __restrict__ r_r,
                                 const float* __restrict__ o_r,
                                 const float* __restrict__ W,
                                 float* __restrict__ Reff) {
  const int lane = threadIdx.x & 31;
  const int wave = threadIdx.x >> 5;
  const int tile = blockIdx.x * 8 + wave;
  const int ti = tile & 127;                // col tile (i)
  const int tj = tile >> 7;                 // row tile (j)
  const int l16 = lane & 15, hi = lane >> 4;
  const int j = tj * 16 + l16;              // A row
  const int i = ti * 16 + l16;              // B col
  v8f acc;
#pragma unroll
  for (int r = 0; r < 8; ++r)
    acc[r] = r_r[(size_t)(tj * 16 + r + 8 * hi) * RSIZE + i];
  for (int kb = 0; kb < OUT_DIM; kb += 4) {
    const int ka = kb + 2 * hi;
    v2f a, bb;
    a.x = o_r[ka * RSIZE + j];              // A = o_r^T
    a.y = o_r[(ka + 1) * RSIZE + j];
    bb.x = W[ka * RSIZE + i];               // B = W
    bb.y = W[(ka + 1) * RSIZE + i];
    acc = wmma_f32_16x16x4(a, bb, acc);
  }
#pragma unroll
  for (int r = 0; r < 8; ++r)
    Reff[(size_t)(tj * 16 + r + 8 * hi) * RSIZE + i] = acc[r];
}

// ---------------------------------------------------------------------------
// Out[T, OUT] = S[T, RSIZE] @ W^T + b       (512 x 8 tiles)
__global__ void gemm_out_kernel(const float* __restrict__ S,
                                const float* __restrict__ W,
                                const float* __restrict__ b,
                                float* __restrict__ Out) {
  const int lane = threadIdx.x & 31;
  const int wave = threadIdx.x >> 5;
  const int tile = blockIdx.x * 8 + wave;
  const int to = tile & 7;                  // col tile (out dim)
  const int tt = tile >> 3;                 // row tile (time)
  const int l16 = lane & 15, hi = lane >> 4;
  const int m = tt * 16 + l16;              // time row
  const int n = to * 16 + l16;              // output col
  const float bn = b[n];
  v8f acc;
#pragma unroll
  for (int r = 0; r < 8; ++r) acc[r] = bn;
  for (int kb = 0; kb < RSIZE; kb += 4) {
    const int ka = kb + 2 * hi;
    v2f a, bb;
    a.x = S[(size_t)m * RSIZE + ka];
    a.y = S[(size_t)m * RSIZE + ka + 1];
    bb.x = W[(size_t)n * RSIZE + ka];       // B = W^T
    bb.y = W[(size_t)n * RSIZE + ka + 1];
    acc = wmma_f32_16x16x4(a, bb, acc);
  }
#pragma unroll
  for (int r = 0; r < 8; ++r)
    Out[(size_t)(tt * 16 + r + 8 * hi) * OUT_DIM + n] = acc[r];
}

// ---------------------------------------------------------------------------
__device__ __forceinline__ void gridBarrier(unsigned* cnt, unsigned* gen) {
  __syncthreads();
  if (threadIdx.x == 0) {
    __threadfence();
    unsigned g = __atomic_load_n(gen, __ATOMIC_ACQUIRE);
    if (__atomic_fetch_add(cnt, 1u, __ATOMIC_ACQ_REL) == (unsigned)(NB - 1)) {
      __atomic_store_n(cnt, 0u, __ATOMIC_RELAXED);
      __atomic_fetch_add(gen, 1u, __ATOMIC_ACQ_REL);   // release next phase
    } else {
      while (__atomic_load_n(gen, __ATOMIC_ACQUIRE) == g)
        __builtin_amdgcn_s_sleep(2);
    }
    __threadfence();
  }
  __syncthreads();
}

// Persistent serial-recurrence kernel. Block b owns rows [16b, 16b+16) of
// R_eff, kept in LDS (128 KB). 16 threads cooperate per row.
__global__ void esn_recurrent_kernel(const float* __restrict__ U,
                                     const float* __restrict__ Reff,
                                     const float* __restrict__ cvec,
                                     float* __restrict__ S,
                                     float* __restrict__ sbuf,   // 2 x RSIZE
                                     unsigned* __restrict__ bar) {
  extern __shared__ float lds[];
  float* ldsR = lds;                        // ROWS_PB * RSIZE
  float* ldsS = lds + ROWS_PB * RSIZE;      // RSIZE
  const int blk = blockIdx.x;
  const int tid = threadIdx.x;

  // Preload this block's R_eff rows (contiguous 128 KB, coalesced).
  const size_t base = (size_t)blk * ROWS_PB * RSIZE;
  for (int i = tid; i < ROWS_PB * RSIZE; i += 256) ldsR[i] = Reff[base + i];

  const int r   = tid >> 4;                 // local row 0..15
  const int l16 = tid & 15;                 // lane-in-row
  const int gi  = blk * ROWS_PB + r;        // global reservoir row
  __syncthreads();

  for (int t = 0; t < T_STEPS; ++t) {
    float pre;
    if (t == 0) {
      pre = U[gi];                          // state=0, out=0 at t=0
    } else {
      // Stage state(t-1) into LDS.
      const float* sin = sbuf + ((t + 1) & 1) * RSIZE;
      for (int i = tid; i < RSIZE; i += 256) ldsS[i] = sin[i];
      __syncthreads();
      const float* rrow = ldsR + r * RSIZE;
      float ps = 0.f;
#pragma unroll 8
      for (int k = 0; k < RSIZE / 16; ++k) {
        const int j = l16 + (k << 4);
        ps = fmaf(rrow[j], ldsS[j], ps);
      }
      // Reduce across the 16-lane row group (wave32-safe xor butterflies).
      for (int off = 8; off; off >>= 1) ps += __shfl_xor(ps, off, 32);
      pre = U[(size_t)t * RSIZE + gi] + cvec[gi] + ps;
    }
    const float sv = tanhf(pre);
    if (l16 == 0) {
      sbuf[(t & 1) * RSIZE + gi] = sv;      // double-buffered live state
      S[(size_t)t * RSIZE + gi] = sv;       // trajectory for final GEMM
    }
    if (tid == 0 && t + 1 < T_STEPS)        // pull next step's U rows into L2
      __builtin_prefetch(&U[(size_t)(t + 1) * RSIZE + blk * ROWS_PB], 0, 0);
    gridBarrier(bar, bar + 1);
  }
}

// ---------------------------------------------------------------------------
extern "C" void kernel_launch(void* const* d_in, const int* in_sizes, int n_in,
                              void* d_out, int out_size, void* d_ws, size_t ws_size,
                              hipStream_t stream) {
  const float* X   = (const float*)d_in[0];   // [T,1,IN]
  const float* i_r = (const float*)d_in[1];   // [IN, RSIZE]
  const float* r_r = (const float*)d_in[2];   // [RSIZE, RSIZE]
  const float* o_r = (const float*)d_in[3];   // [OUT, RSIZE]
  const float* W   = (const float*)d_in[4];   // [OUT, RSIZE]
  const float* b   = (const float*)d_in[5];   // [OUT]
  float* out = (float*)d_out;                 // [T,1,OUT]

  float* ws = (float*)d_ws;
  size_t off = 0;
  float* U    = ws + off; off += (size_t)T_STEPS * RSIZE;   // 64 MB
  float* S    = ws + off; off += (size_t)T_STEPS * RSIZE;   // 64 MB
  float* Reff = ws + off; off += (size_t)RSIZE * RSIZE;     // 16 MB
  float* cvec = ws + off; off += RSIZE;
  float* sbuf = ws + off; off += 2 * RSIZE;
  unsigned* bar = (unsigned*)(ws + off);

  init_barrier_kernel<<<1, 1, 0, stream>>>(bar);
  c_vec_kernel<<<RSIZE / 256, 256, 0, stream>>>(o_r, b, cvec);
  gemm_u_kernel<<<(T_STEPS / 16) * (RSIZE / 16) / 8, 256, 0, stream>>>(X, i_r, U);
  gemm_reff_kernel<<<(RSIZE / 16) * (RSIZE / 16) / 8, 256, 0, stream>>>(r_r, o_r, W, Reff);

  const size_t ldsBytes = (size_t)(ROWS_PB * RSIZE + RSIZE) * sizeof(float); // 136 KB
  esn_recurrent_kernel<<<NB, 256, ldsBytes, stream>>>(U, Reff, cvec, S, sbuf, bar);

  gemm_out_kernel<<<(T_STEPS / 16) * (OUT_DIM / 16) / 8, 256, 0, stream>>>(S, W, b, out);
}